// ICP_77773267796662
// MI455X (gfx1250) — compile-verified
//
#include <hip/hip_runtime.h>
#include <hip/hip_bf16.h>
#include <float.h>
#include <math.h>

// ---------------------------------------------------------------------------
// ICP (B=1, N=4096, 10 steps) for gfx1250.
//
// NN core: dist^2(i,j) = |pc_i|^2 + |q_j|^2 - 2 pc_i.q_j ; argmin_j only needs
// (|q_j|^2 - 2 dot). dot computed 16x16 tiles at a time with
// V_WMMA_F32_16X16X4_F32 (K=3 padded to 4 with zeros). Column sweep unrolled
// 4x so B-fragment loads overlap the WMMA+argmin VALU work of prior tiles.
// ---------------------------------------------------------------------------

typedef __attribute__((ext_vector_type(2))) float v2f;
typedef __attribute__((ext_vector_type(8))) float v8f;

#define N_PTS   4096
#define N_TILES (N_PTS / 16)
#define UNROLL  4

// Workspace layout (bytes):
//   pc4   : float4[4096] @ 0       (65536)  current (moving) cloud, xyz0
//   q4    : float4[4096] @ 65536   (65536)  target cloud, xyz0
//   qnorm : float [4096] @ 131072  (16384)  |q|^2
//   idx   : int   [4096] @ 147456  (16384)  NN indices for current step

__global__ void icp_prep(const float* __restrict__ p1, const float* __restrict__ p2,
                         float4* __restrict__ pc4, float4* __restrict__ q4,
                         float* __restrict__ qnorm) {
  int i = blockIdx.x * blockDim.x + threadIdx.x;
  if (i >= N_PTS) return;
  float x1 = p1[3 * i + 0], y1 = p1[3 * i + 1], z1 = p1[3 * i + 2];
  pc4[i] = make_float4(x1, y1, z1, 0.0f);
  float x2 = p2[3 * i + 0], y2 = p2[3 * i + 1], z2 = p2[3 * i + 2];
  q4[i] = make_float4(x2, y2, z2, 0.0f);
  qnorm[i] = x2 * x2 + y2 * y2 + z2 * z2;
}

// One wave per 16-row tile; 2 waves per block, 128 blocks -> 256 waves = 256 tiles.
__global__ void icp_nn(const float4* __restrict__ pc4, const float4* __restrict__ q4,
                       const float* __restrict__ qnorm, int* __restrict__ idxOut) {
  const int gtid = blockIdx.x * blockDim.x + threadIdx.x;
  const int wave = gtid >> 5;
  const int lane = threadIdx.x & 31;
  const int m    = lane & 15;   // point-in-tile index this lane covers
  const int hi   = lane >> 4;   // 0: coords (x,y) / K=0,1 ; 1: coords (z,0) / K=2,3
  const int row0 = wave * 16;

  // A fragment: 16x4 f32 (ISA layout: lanes 0-15 K=0,1 ; lanes 16-31 K=2,3)
  const float* ap = reinterpret_cast<const float*>(pc4 + (row0 + m));
  v2f a;
  a.x = ap[2 * hi + 0];
  a.y = ap[2 * hi + 1];

  float minv[8];
  int   mini[8];
#pragma unroll
  for (int r = 0; r < 8; ++r) { minv[r] = FLT_MAX; mini[r] = 0; }

  for (int j = 0; j < N_TILES; j += UNROLL) {
    v2f   b[UNROLL];
    float nb[UNROLL];
    // Batch the B-fragment + norm loads so they overlap the WMMA/VALU below.
#pragma unroll
    for (int u = 0; u < UNROLL; ++u) {
      const int    col = (j + u) * 16 + m;
      const float* bp  = reinterpret_cast<const float*>(q4 + col);
      b[u].x = bp[2 * hi + 0];
      b[u].y = bp[2 * hi + 1];
      nb[u]  = qnorm[col];
    }
#pragma unroll
    for (int u = 0; u < UNROLL; ++u) {
      v8f c = {};
      // D = A x B (+0) : 16x16 dot products pc_row . q_col
      c = __builtin_amdgcn_wmma_f32_16x16x4_f32(false, a, false, b[u], (short)0, c,
                                                false, false);
      // C layout: VGPR r, lanes 0-15 -> (M=r, N=lane); lanes 16-31 -> (M=r+8, N=lane-16)
      const int n = (j + u) * 16 + m;
#pragma unroll
      for (int r = 0; r < 8; ++r) {
        float score = fmaf(-2.0f, c[r], nb[u]);  // |q_n|^2 - 2 dot
        if (score < minv[r]) { minv[r] = score; mini[r] = n; }
      }
    }
  }

  // Per-row min across the 16 lanes of each half-wave; tie-break = lowest index
  // (matches jnp.argmin first-occurrence semantics).
#pragma unroll
  for (int r = 0; r < 8; ++r) {
    float v  = minv[r];
    int   id = mini[r];
#pragma unroll
    for (int off = 8; off >= 1; off >>= 1) {
      float ov = __shfl_xor(v, off, 32);
      int   oi = __shfl_xor(id, off, 32);
      if (ov < v || (ov == v && oi < id)) { v = ov; id = oi; }
    }
    if (m == 0) idxOut[row0 + r + 8 * hi] = id;  // row M = r + 8*hi
  }
}

__device__ __forceinline__ float det3(const float M[9]) {
  return M[0] * (M[4] * M[8] - M[5] * M[7])
       - M[1] * (M[3] * M[8] - M[5] * M[6])
       + M[2] * (M[3] * M[7] - M[4] * M[6]);
}

// mode 0: src = pc, tgt = q[idx]; compute (R,t) and update pc in place.
// mode 1: src = p1, tgt = pc;     compute (R,t) and write [3x4] (R|t) to out.
__global__ void icp_fit(const float* __restrict__ p1, float4* __restrict__ pc4,
                        const float4* __restrict__ q4, const int* __restrict__ idx,
                        float* __restrict__ out, int mode) {
  const int tid = threadIdx.x;
  __shared__ float red[256];
  __shared__ float sums[15];
  __shared__ float Rt[12];

  float acc[15];
#pragma unroll
  for (int c = 0; c < 15; ++c) acc[c] = 0.0f;

  for (int k = 0; k < N_PTS / 256; ++k) {
    int i = tid + 256 * k;
    float sx, sy, sz, tx, ty, tz;
    if (mode == 0) {
      float4 p = pc4[i];
      sx = p.x; sy = p.y; sz = p.z;
      float4 q = q4[idx[i]];
      tx = q.x; ty = q.y; tz = q.z;
    } else {
      sx = p1[3 * i + 0]; sy = p1[3 * i + 1]; sz = p1[3 * i + 2];
      float4 p = pc4[i];
      tx = p.x; ty = p.y; tz = p.z;
    }
    acc[0] += sx; acc[1] += sy; acc[2] += sz;
    acc[3] += tx; acc[4] += ty; acc[5] += tz;
    acc[6]  += sx * tx; acc[7]  += sx * ty; acc[8]  += sx * tz;
    acc[9]  += sy * tx; acc[10] += sy * ty; acc[11] += sy * tz;
    acc[12] += sz * tx; acc[13] += sz * ty; acc[14] += sz * tz;
  }

  for (int c = 0; c < 15; ++c) {
    red[tid] = acc[c];
    __syncthreads();
    for (int s = 128; s > 0; s >>= 1) {
      if (tid < s) red[tid] += red[tid + s];
      __syncthreads();
    }
    if (tid == 0) sums[c] = red[0];
    __syncthreads();
  }

  if (tid == 0) {
    const float invN = 1.0f / (float)N_PTS;
    float c1[3] = { sums[0] * invN, sums[1] * invN, sums[2] * invN };
    float c2[3] = { sums[3] * invN, sums[4] * invN, sums[5] * invN };
    // H = sum (s - c1)(t - c2)^T = sum s t^T - N c1 c2^T
    float H[9];
    for (int i = 0; i < 3; ++i)
      for (int j = 0; j < 3; ++j)
        H[i * 3 + j] = sums[6 + i * 3 + j] - (float)N_PTS * c1[i] * c2[j];

    // One-sided Jacobi SVD: H = U S V^T
    float A[9];
    for (int i = 0; i < 9; ++i) A[i] = H[i];
    float V[9] = {1, 0, 0, 0, 1, 0, 0, 0, 1};
    for (int sweep = 0; sweep < 15; ++sweep) {
      for (int pp = 0; pp < 3; ++pp) {
        const int p = (pp == 2) ? 1 : 0;
        const int q = (pp == 0) ? 1 : 2;
        float app = 0.f, aqq = 0.f, apq = 0.f;
        for (int i = 0; i < 3; ++i) {
          float x = A[i * 3 + p], y = A[i * 3 + q];
          app += x * x; aqq += y * y; apq += x * y;
        }
        if (fabsf(apq) < 1e-20f) continue;
        float zeta = (aqq - app) / (2.0f * apq);
        float tth  = copysignf(1.0f, zeta) / (fabsf(zeta) + sqrtf(1.0f + zeta * zeta));
        float cc   = 1.0f / sqrtf(1.0f + tth * tth);
        float ss   = cc * tth;
        for (int i = 0; i < 3; ++i) {
          float x = A[i * 3 + p], y = A[i * 3 + q];
          A[i * 3 + p] = cc * x - ss * y;
          A[i * 3 + q] = ss * x + cc * y;
          x = V[i * 3 + p]; y = V[i * 3 + q];
          V[i * 3 + p] = cc * x - ss * y;
          V[i * 3 + q] = ss * x + cc * y;
        }
      }
    }
    float sig[3], U[9];
    for (int k = 0; k < 3; ++k) {
      float s2 = 0.f;
      for (int i = 0; i < 3; ++i) s2 += A[i * 3 + k] * A[i * 3 + k];
      sig[k] = sqrtf(s2);
    }
    // sort singular values descending (match jnp.linalg.svd ordering)
    for (int a0 = 0; a0 < 2; ++a0)
      for (int b0 = 0; b0 < 2 - a0; ++b0)
        if (sig[b0] < sig[b0 + 1]) {
          float tmp = sig[b0]; sig[b0] = sig[b0 + 1]; sig[b0 + 1] = tmp;
          for (int i = 0; i < 3; ++i) {
            float ta = A[i * 3 + b0]; A[i * 3 + b0] = A[i * 3 + b0 + 1]; A[i * 3 + b0 + 1] = ta;
            float tv = V[i * 3 + b0]; V[i * 3 + b0] = V[i * 3 + b0 + 1]; V[i * 3 + b0 + 1] = tv;
          }
        }
    for (int k = 0; k < 3; ++k) {
      float inv = 1.0f / fmaxf(sig[k], 1e-20f);
      for (int i = 0; i < 3; ++i) U[i * 3 + k] = A[i * 3 + k] * inv;
    }
    // reflection fix: R = V diag(1,1,det(V U^T)) U^T
    float d  = det3(V) * det3(U);
    float dk = (d < 0.0f) ? -1.0f : 1.0f;
    float R[9];
    for (int i = 0; i < 3; ++i)
      for (int j = 0; j < 3; ++j)
        R[i * 3 + j] = V[i * 3 + 0] * U[j * 3 + 0]
                     + V[i * 3 + 1] * U[j * 3 + 1]
                     + dk * V[i * 3 + 2] * U[j * 3 + 2];
    float t[3];
    for (int i = 0; i < 3; ++i)
      t[i] = c2[i] - (R[i * 3 + 0] * c1[0] + R[i * 3 + 1] * c1[1] + R[i * 3 + 2] * c1[2]);
    for (int i = 0; i < 9; ++i) Rt[i] = R[i];
    Rt[9] = t[0]; Rt[10] = t[1]; Rt[11] = t[2];
  }
  __syncthreads();

  if (mode == 0) {
    const float R0 = Rt[0], R1 = Rt[1], R2 = Rt[2];
    const float R3 = Rt[3], R4 = Rt[4], R5 = Rt[5];
    const float R6 = Rt[6], R7 = Rt[7], R8 = Rt[8];
    const float t0 = Rt[9], t1 = Rt[10], t2 = Rt[11];
    for (int k = 0; k < N_PTS / 256; ++k) {
      int i = tid + 256 * k;
      float4 p = pc4[i];
      float nx = R0 * p.x + R1 * p.y + R2 * p.z + t0;
      float ny = R3 * p.x + R4 * p.y + R5 * p.z + t1;
      float nz = R6 * p.x + R7 * p.y + R8 * p.z + t2;
      pc4[i] = make_float4(nx, ny, nz, 0.0f);
    }
  } else {
    if (tid < 12) {
      int i = tid >> 2, j = tid & 3;  // out[3][4] = (R | t), row-major
      out[tid] = (j < 3) ? Rt[i * 3 + j] : Rt[9 + i];
    }
  }
}

extern "C" void kernel_launch(void* const* d_in, const int* in_sizes, int n_in,
                              void* d_out, int out_size, void* d_ws, size_t ws_size,
                              hipStream_t stream) {
  (void)in_sizes; (void)n_in; (void)out_size; (void)ws_size;
  const float* p1 = (const float*)d_in[0];
  const float* p2 = (const float*)d_in[1];
  char* ws = (char*)d_ws;
  float4* pc4   = (float4*)(ws + 0);
  float4* q4    = (float4*)(ws + 65536);
  float*  qnorm = (float*)(ws + 131072);
  int*    idx   = (int*)(ws + 147456);
  float*  out   = (float*)d_out;

  icp_prep<<<16, 256, 0, stream>>>(p1, p2, pc4, q4, qnorm);
  for (int step = 0; step < 10; ++step) {
    // 256 row-tiles, one wave each; 64-thread blocks spread waves over more WGPs
    icp_nn<<<128, 64, 0, stream>>>(pc4, q4, qnorm, idx);
    icp_fit<<<1, 256, 0, stream>>>(p1, pc4, q4, idx, out, /*mode=*/0);
  }
  icp_fit<<<1, 256, 0, stream>>>(p1, pc4, q4, idx, out, /*mode=*/1);
}